// DiffFDN_21079699489094
// MI455X (gfx1250) — compile-verified
//
#include <hip/hip_runtime.h>
#include <hip/hip_bf16.h>
#include <stdint.h>

// ---------------------------------------------------------------------------
// DiffFDN on MI455X (gfx1250, wave32).
//   K1 prep:   MLP projections + 8x8 skew matrix exponentials (LDS, VALU)
//   K2 chain:  Uz = D U1 D U2 D U3 D U4 per (b,f).  Complex 8x8 state kept as
//              real [Re;Im] 16x8 stack, 2 bins packed in columns -> exact
//              16x16x16 f32 matmuls via v_wmma_f32_16x16x4_f32.  D-diagonal
//              applied as per-row rotation using ds_swizzle SWAPX16.
//              Uz written as float2 (b64) stores, Re/Im paired via swizzle.
//   K3 solve:  t2 = inv(I - Uz*Gamma) * (Uz*Gamma*B); H = C.t2  (register GJ,
//              b128 loads of Uz), writes interleaved {cos,sin} DFT coeffs.
//   K4 idft:   irfft as 16xK GEMM with WMMA; unroll x2, dual per-lane phasor
//              streams, b64 coefficient loads, global_prefetch ahead.
// ---------------------------------------------------------------------------

#define NFFT   48000
#define FBINS  24001
#define NB     16
#define NDEL   8
#define NSTAGE 4
#define TWO_PI_OVER_N (6.28318530717958647692f / 48000.0f)

typedef __attribute__((ext_vector_type(2))) float v2f;
typedef __attribute__((ext_vector_type(8))) float v8f;

__device__ __forceinline__ v8f wmma4(v2f a, v2f b, v8f c) {
  // D(16x16 f32) = A(16x4 f32) * B(4x16 f32) + C
  return __builtin_amdgcn_wmma_f32_16x16x4_f32(
      false, a, false, b, (short)0, c, false, false);
}

__device__ __forceinline__ float swapx16(float x) {
  // group-of-32 swizzle, xor_mask=0x10 (SWAPX16): lane <-> lane^16
  return __int_as_float(
      __builtin_amdgcn_ds_swizzle(__float_as_int(x), 0x401F));
}

// ---------------------------------------------------------------------------
// Kernel 1: per-batch MLP projections + expm + gamma powers
// ---------------------------------------------------------------------------
template <int O, int C>
__device__ void proj(const float* __restrict__ xb, const float* __restrict__ w1,
                     const float* __restrict__ b1, const float* __restrict__ w2,
                     const float* __restrict__ b2, float* t_lds, float* out_lds) {
  const int tid = threadIdx.x;  // 256 threads, tid == s
  float acc[O];
#pragma unroll
  for (int o = 0; o < O; ++o) acc[o] = 0.f;
  for (int f = 0; f < 84; ++f) {
    float xv = xb[f * 256 + tid];
#pragma unroll
    for (int o = 0; o < O; ++o) acc[o] = fmaf(w1[o * 84 + f], xv, acc[o]);
  }
#pragma unroll
  for (int o = 0; o < O; ++o) t_lds[o * 256 + tid] = acc[o] + b1[o];
  __syncthreads();
  if (tid < O * C) {
    int o = tid / C, c = tid % C;
    float s = 0.f;
    for (int ss = 0; ss < 256; ++ss)
      s = fmaf(t_lds[o * 256 + ss], w2[c * 256 + ss], s);
    out_lds[o * 8 + c] = s + b2[c];
  }
  __syncthreads();
}

__device__ void expm8(const float* sM, float* sA, float* sP) {
  // skew(M) then scaling(2^-7)+Taylor(8)+squaring(7); 64 threads cooperate
  const int tid = threadIdx.x;
  const int i = (tid & 63) >> 3, j = tid & 7;
  if (tid < 64) {
    float m = 0.f;
    if (j > i) m = sM[i * 8 + j];
    else if (j < i) m = -sM[j * 8 + i];
    sA[tid] = m * 0.0078125f;
    sP[tid] = (i == j) ? 1.f : 0.f;
  }
  __syncthreads();
  for (int n = 8; n >= 1; --n) {
    float r = 0.f;
    if (tid < 64) {
#pragma unroll
      for (int l = 0; l < 8; ++l) r = fmaf(sA[i * 8 + l], sP[l * 8 + j], r);
    }
    __syncthreads();
    if (tid < 64) sP[tid] = ((i == j) ? 1.f : 0.f) + r * (1.0f / (float)n);
    __syncthreads();
  }
  for (int q = 0; q < 7; ++q) {
    float r = 0.f;
    if (tid < 64) {
#pragma unroll
      for (int l = 0; l < 8; ++l) r = fmaf(sP[i * 8 + l], sP[l * 8 + j], r);
    }
    __syncthreads();
    if (tid < 64) sP[tid] = r;
    __syncthreads();
  }
}

__global__ __launch_bounds__(256) void prep_kernel(
    const float* __restrict__ x,
    const float* __restrict__ Uw1, const float* __restrict__ Ub1,
    const float* __restrict__ Uw2, const float* __restrict__ Ub2,
    const float* __restrict__ Gw1, const float* __restrict__ Gb1,
    const float* __restrict__ Gw2, const float* __restrict__ Gb2,
    const float* __restrict__ BCw1, const float* __restrict__ BCb1,
    const float* __restrict__ BCw2, const float* __restrict__ BCb2,
    const int* __restrict__ delays,
    float* __restrict__ wsU, float* __restrict__ wsGd,
    float* __restrict__ wsGB, float* __restrict__ wsCv,
    float* __restrict__ ab) {
  __shared__ float t_lds[8 * 256];
  __shared__ float sM[64], sA[64], sP[64];
  __shared__ float sg;
  const int b = blockIdx.x, tid = threadIdx.x;
  const float* xb = x + (size_t)b * 84 * 256;

  for (int k = 0; k < NSTAGE; ++k) {
    proj<8, 8>(xb, Uw1 + k * 8 * 84, Ub1 + k * 8, Uw2 + k * 8 * 256, Ub2 + k * 8,
               t_lds, sM);
    expm8(sM, sA, sP);
    if (tid < 64) wsU[(b * NSTAGE + k) * 64 + tid] = sP[tid];
    __syncthreads();
  }
  proj<1, 1>(xb, Gw1, Gb1, Gw2, Gb2, t_lds, sM);
  if (tid == 0) sg = sM[0];
  __syncthreads();
  proj<2, 8>(xb, BCw1, BCb1, BCw2, BCb2, t_lds, sM);
  if (tid < 8) {
    int m = delays[tid];
    float g = fmaxf(sg, 1e-12f);
    float gd = __powf(g, (float)m);
    wsGd[b * 8 + tid] = gd;
    wsGB[b * 8 + tid] = gd * sM[0 * 8 + tid];  // Gamma @ B
    wsCv[b * 8 + tid] = sM[1 * 8 + tid];       // C
  }
  // zero pad row (f == FBINS) of interleaved DFT coefficient table
  if (b == 0 && tid < 32) ab[(size_t)FBINS * 32 + tid] = 0.f;
}

// ---------------------------------------------------------------------------
// Kernel 2: Uz chain via WMMA.  One wave handles (b, f0, f1=f0+1).
// State S (16x16 f32, C/D layout): rows 0-7 = Re, 8-15 = Im;
// cols 0-7 = bin f0, cols 8-15 = bin f1.
// ---------------------------------------------------------------------------
__global__ __launch_bounds__(256) void chain_kernel(
    const float* __restrict__ wsU, const int* __restrict__ delays,
    float* __restrict__ uz_out) {
  const int wid = blockIdx.x * 8 + (threadIdx.x >> 5);
  const int PAIRS = (FBINS + 1) / 2;  // 12001
  const int b = wid / PAIRS;
  const int fp = wid - b * PAIRS;
  const int f0 = fp * 2;
  const int f1 = (f0 + 1 < FBINS) ? f0 + 1 : f0;
  const int L = threadIdx.x & 31;
  const int Ncol = L & 15;
  const bool hiHalf = (L >= 16);
  const bool lo = !hiHalf;
  const int colj = Ncol & 7;
  const int fmine = (Ncol < 8) ? f0 : f1;
  const int Mrow = L & 15;

  // A operands: blockdiag(Uk, Uk) in WMMA-A layout (constant across all f)
  v2f Amat[3][4];
#pragma unroll
  for (int k = 0; k < 3; ++k) {
#pragma unroll
    for (int s = 0; s < 4; ++s) {
#pragma unroll
      for (int vi = 0; vi < 2; ++vi) {
        int Kc = s * 4 + vi + (hiHalf ? 2 : 0);
        float v = 0.f;
        if ((Mrow < 8) == (Kc < 8))
          v = wsU[(b * NSTAGE + k) * 64 + (Mrow & 7) * 8 + (Kc & 7)];
        Amat[k][s][vi] = v;
      }
    }
  }

  // per-row rotation coeffs for D = diag(z^-m): theta = 2*pi*f*m_j/NFFT
  float cc[8], ssg[8];
#pragma unroll
  for (int r = 0; r < 8; ++r) {
    unsigned m = (unsigned)delays[r];
    unsigned units = ((unsigned)fmine * m) % (unsigned)NFFT;
    float sn, cs;
    __sincosf((float)units * TWO_PI_OVER_N, &sn, &cs);
    cc[r] = cs;
    ssg[r] = hiHalf ? -sn : sn;  // Re' = c*Re + s*Im ; Im' = c*Im - s*Re
  }

  // init S = [Re = U_{K-1}; Im = 0]
  v8f S;
#pragma unroll
  for (int r = 0; r < 8; ++r)
    S[r] = hiHalf ? 0.f : wsU[(b * NSTAGE + 3) * 64 + r * 8 + colj];

#pragma unroll
  for (int k = 2; k >= 0; --k) {
    // rotate: S = D * S (row-wise complex rotation across half-waves)
#pragma unroll
    for (int r = 0; r < 8; ++r) {
      float sw = swapx16(S[r]);
      S[r] = fmaf(cc[r], S[r], ssg[r] * sw);
    }
    // D-layout -> B-layout (rows 4s..4s+3 per slice)
    float sz0 = swapx16(S[0]), sz1 = swapx16(S[1]), sz2 = swapx16(S[2]),
          sz3 = swapx16(S[3]), sz4 = swapx16(S[4]), sz5 = swapx16(S[5]),
          sz6 = swapx16(S[6]), sz7 = swapx16(S[7]);
    v2f B0 = {lo ? S[0] : sz2, lo ? S[1] : sz3};
    v2f B1 = {lo ? S[4] : sz6, lo ? S[5] : sz7};
    v2f B2 = {lo ? sz0 : S[2], lo ? sz1 : S[3]};
    v2f B3 = {lo ? sz4 : S[6], lo ? sz5 : S[7]};
    v8f D = {0, 0, 0, 0, 0, 0, 0, 0};
    D = wmma4(Amat[k][0], B0, D);
    D = wmma4(Amat[k][1], B1, D);
    D = wmma4(Amat[k][2], B2, D);
    D = wmma4(Amat[k][3], B3, D);
    S = D;
  }
  // final leftmost D
#pragma unroll
  for (int r = 0; r < 8; ++r) {
    float sw = swapx16(S[r]);
    S[r] = fmaf(cc[r], S[r], ssg[r] * sw);
  }

  // Pair Re/Im across half-waves (partner lane L^16 holds the other part of
  // the same (row,col)); swizzles before predicate so EXEC is full.
  float swA[4], swB[4];
#pragma unroll
  for (int r = 0; r < 4; ++r) {
    swA[r] = swapx16(S[r]);      // lo lanes: Im of row r   | hi: Re of row r
    swB[r] = swapx16(S[r + 4]);  // lo lanes: Im of row r+4 | hi: Re of row r+4
  }
  // store Uz[b, fmine] as complex64: lo lanes store rows 0-3, hi rows 4-7
  const bool okst = (Ncol < 8) || (f0 + 1 < FBINS);
  if (okst) {
    v2f* outc = (v2f*)(uz_out + ((size_t)b * FBINS + (size_t)fmine) * 128);
#pragma unroll
    for (int r = 0; r < 4; ++r) {
      v2f val;
      val.x = lo ? S[r] : swB[r];      // Re
      val.y = lo ? swA[r] : S[r + 4];  // Im
      const int rr = lo ? r : (r + 4);
      outc[rr * 8 + colj] = val;       // global_store_b64
    }
  }
}

// ---------------------------------------------------------------------------
// Kernel 3: per-(b,f) complex 8x8 solve (Gauss-Jordan, fully unrolled regs)
// ---------------------------------------------------------------------------
__global__ __launch_bounds__(256) void solve_kernel(
    const float* __restrict__ uz, const float* __restrict__ wsGd,
    const float* __restrict__ wsGB, const float* __restrict__ wsCv,
    v2f* __restrict__ ab) {
  const int idx = blockIdx.x * 256 + threadIdx.x;
  if (idx >= NB * FBINS) return;
  const int b = idx / FBINS;
  const int f = idx - b * FBINS;
  const float4* M4 = (const float4*)(uz + ((size_t)b * FBINS + (size_t)f) * 128);

  float gd[8], gb[8], cv[8];
#pragma unroll
  for (int j = 0; j < 8; ++j) {
    gd[j] = wsGd[b * 8 + j];
    gb[j] = wsGB[b * 8 + j];
    cv[j] = wsCv[b * 8 + j];
  }
  float Ar[8][8], Ai[8][8], tr[8], ti[8];
#pragma unroll
  for (int i = 0; i < 8; ++i) {
    tr[i] = 0.f; ti[i] = 0.f;
#pragma unroll
    for (int jp = 0; jp < 4; ++jp) {      // b128: two complex entries at once
      float4 q = M4[i * 4 + jp];
      const int j0 = jp * 2, j1 = jp * 2 + 1;
      tr[i] = fmaf(q.x, gb[j0], tr[i]);   // t1 = Uz @ (Gamma B)
      ti[i] = fmaf(q.y, gb[j0], ti[i]);
      Ar[i][j0] = ((i == j0) ? 1.f : 0.f) - q.x * gd[j0];  // I - Uz*Gamma
      Ai[i][j0] = -q.y * gd[j0];
      tr[i] = fmaf(q.z, gb[j1], tr[i]);
      ti[i] = fmaf(q.w, gb[j1], ti[i]);
      Ar[i][j1] = ((i == j1) ? 1.f : 0.f) - q.z * gd[j1];
      Ai[i][j1] = -q.w * gd[j1];
    }
  }
  // Gauss-Jordan (A diagonally dominant-ish; no pivoting)
#pragma unroll
  for (int k = 0; k < 8; ++k) {
    float dr = Ar[k][k], di = Ai[k][k];
    float inv = 1.f / (dr * dr + di * di + 1e-30f);
    float pr = dr * inv, pi = -di * inv;  // 1/A[k][k]
#pragma unroll
    for (int j = 0; j < 8; ++j) {
      float xr = Ar[k][j], xi = Ai[k][j];
      Ar[k][j] = xr * pr - xi * pi;
      Ai[k][j] = xr * pi + xi * pr;
    }
    {
      float xr = tr[k], xi = ti[k];
      tr[k] = xr * pr - xi * pi;
      ti[k] = xr * pi + xi * pr;
    }
#pragma unroll
    for (int i = 0; i < 8; ++i) {
      if (i == k) continue;
      float fr = Ar[i][k], fi = Ai[i][k];
#pragma unroll
      for (int j = 0; j < 8; ++j) {
        Ar[i][j] -= fr * Ar[k][j] - fi * Ai[k][j];
        Ai[i][j] -= fr * Ai[k][j] + fi * Ar[k][j];
      }
      tr[i] -= fr * tr[k] - fi * ti[k];
      ti[i] -= fr * ti[k] + fi * tr[k];
    }
  }
  float hr = 0.f, hi = 0.f;
#pragma unroll
  for (int i = 0; i < 8; ++i) {
    hr = fmaf(cv[i], tr[i], hr);
    hi = fmaf(cv[i], ti[i], hi);
  }
  const bool edge = (f == 0) || (f == FBINS - 1);
  const float w = edge ? (1.f / (float)NFFT) : (2.f / (float)NFFT);
  v2f coeff;
  coeff.x = w * hr;                 // cos-row coefficient
  coeff.y = edge ? 0.f : -w * hi;   // sin-row coefficient
  ab[(size_t)f * 16 + b] = coeff;   // interleaved, b64 store
}

// ---------------------------------------------------------------------------
// Kernel 4: irfft as WMMA GEMM.  One wave per 16 output samples; K loop over
// frequency, 2 wmma/iter with independent phasor streams (each advances 4
// bins), b64 coefficient loads, prefetch ahead, periodic exact resync.
// ---------------------------------------------------------------------------
__global__ __launch_bounds__(256) void idft_kernel(
    const v2f* __restrict__ ab, float* __restrict__ ir) {
  const int w = blockIdx.x * 8 + (threadIdx.x >> 5);
  if (w >= NFFT / 16) return;  // wave-uniform
  const int L = threadIdx.x & 31;
  const int nl = L & 15;
  const int n = w * 16 + nl;
  const unsigned foff = (L >= 16) ? 1u : 0u;
  unsigned u0 = ((unsigned)n * foff) % (unsigned)NFFT;
  unsigned u1 = ((unsigned)n * (foff + 2u)) % (unsigned)NFFT;
  const unsigned du = (4u * (unsigned)n) % (unsigned)NFFT;
  float c0, s0, c1, s1, dc, ds;
  __sincosf((float)u0 * TWO_PI_OVER_N, &s0, &c0);
  __sincosf((float)u1 * TWO_PI_OVER_N, &s1, &c1);
  __sincosf((float)du * TWO_PI_OVER_N, &ds, &dc);

  v8f acc = {0, 0, 0, 0, 0, 0, 0, 0};
  for (int i = 0; i < 6000; ++i) {
    const int f = 4 * i;
    v2f A0 = ab[(size_t)f * 16 + L];        // rows f, f+1 (coalesced b64)
    v2f B0 = {c0, s0};
    acc = wmma4(A0, B0, acc);
    v2f A1 = ab[(size_t)(f + 2) * 16 + L];  // rows f+2, f+3
    v2f B1 = {c1, s1};
    acc = wmma4(A1, B1, acc);
    // advance both streams by 4 bins (independent recurrences)
    float t0 = fmaf(c0, dc, -s0 * ds);
    s0 = fmaf(s0, dc, c0 * ds);
    c0 = t0;
    float t1 = fmaf(c1, dc, -s1 * ds);
    s1 = fmaf(s1, dc, c1 * ds);
    c1 = t1;
    u0 += du; if (u0 >= (unsigned)NFFT) u0 -= (unsigned)NFFT;
    u1 += du; if (u1 >= (unsigned)NFFT) u1 -= (unsigned)NFFT;
    if ((i & 63) == 63) {  // exact resync, kills recurrence drift
      __sincosf((float)u0 * TWO_PI_OVER_N, &s0, &c0);
      __sincosf((float)u1 * TWO_PI_OVER_N, &s1, &c1);
    }
    if ((i & 15) == 0) {   // pull coefficient stream into near caches
      int pf = f + 128;
      if (pf > 24000) pf = 24000;
      __builtin_prefetch((const void*)&ab[(size_t)pf * 16 + L], 0, 0);
    }
  }
  // remainder bin f = 24000 (stream0 is exactly there; hi lanes hit zero pad)
  {
    v2f A0 = ab[(size_t)24000 * 16 + L];
    v2f B0 = {c0, s0};
    acc = wmma4(A0, B0, acc);
  }
#pragma unroll
  for (int r = 0; r < 8; ++r) {
    int bb = r + ((L >= 16) ? 8 : 0);
    ir[(size_t)bb * NFFT + n] = acc[r];
  }
}

// ---------------------------------------------------------------------------
extern "C" void kernel_launch(void* const* d_in, const int* in_sizes, int n_in,
                              void* d_out, int out_size, void* d_ws,
                              size_t ws_size, hipStream_t stream) {
  const float* x    = (const float*)d_in[0];
  const float* Uw1  = (const float*)d_in[2];
  const float* Ub1  = (const float*)d_in[3];
  const float* Uw2  = (const float*)d_in[4];
  const float* Ub2  = (const float*)d_in[5];
  const float* Gw1  = (const float*)d_in[6];
  const float* Gb1  = (const float*)d_in[7];
  const float* Gw2  = (const float*)d_in[8];
  const float* Gb2  = (const float*)d_in[9];
  const float* BCw1 = (const float*)d_in[10];
  const float* BCb1 = (const float*)d_in[11];
  const float* BCw2 = (const float*)d_in[12];
  const float* BCb2 = (const float*)d_in[13];
  const int* delays = (const int*)d_in[15];

  float* out = (float*)d_out;
  float* ir  = out;                          // [16, 48000] float32
  float* uz  = out + (size_t)NB * NFFT;      // [16, 24001, 8, 8] complex64

  float* ws    = (float*)d_ws;
  float* wsU   = ws;                  // 16*4*64  = 4096
  float* wsGd  = ws + 4096;           // 16*8     = 128
  float* wsGB  = ws + 4224;           // 16*8
  float* wsCv  = ws + 4352;           // 16*8
  float* ab    = ws + 4480;           // (FBINS+1)*16 float2 = 768064 floats

  prep_kernel<<<NB, 256, 0, stream>>>(x, Uw1, Ub1, Uw2, Ub2, Gw1, Gb1, Gw2, Gb2,
                                      BCw1, BCb1, BCw2, BCb2, delays, wsU, wsGd,
                                      wsGB, wsCv, ab);
  // 16 * 12001 waves, 8 waves/block
  chain_kernel<<<(NB * ((FBINS + 1) / 2)) / 8, 256, 0, stream>>>(wsU, delays, uz);
  solve_kernel<<<(NB * FBINS + 255) / 256, 256, 0, stream>>>(uz, wsGd, wsGB,
                                                             wsCv, (v2f*)ab);
  idft_kernel<<<(NFFT / 16) / 8, 256, 0, stream>>>((const v2f*)ab, ir);
}